// DistanceCentroidLoss_74603581931673
// MI455X (gfx1250) — compile-verified
//
#include <hip/hip_runtime.h>
#include <hip/hip_bf16.h>

#define MARGIN 5.0f
#define DDIM   128          // embedding dim
#define KCL    64           // clusters
#define WAVES  8
#define BLOCK  256
#define ROWS_PER_BLOCK (WAVES * 16)   // 128 rows / block

typedef float v2f __attribute__((ext_vector_type(2)));
typedef float v8f __attribute__((ext_vector_type(8)));

// ---------------------------------------------------------------------------
// Kernel 0: zero the workspace accumulators (attr[K], rep[K], cnt[K]) + d_out
// ---------------------------------------------------------------------------
__global__ void dcl_init(float* __restrict__ ws, float* __restrict__ out) {
    int t = threadIdx.x;
    if (t < 3 * KCL) ws[t] = 0.0f;
    if (t == 0) out[0] = 0.0f;
}

// ---------------------------------------------------------------------------
// Kernel 1: main fused distance-GEMM + per-cluster accumulation
// ---------------------------------------------------------------------------
__global__ __launch_bounds__(BLOCK)
void dcl_main(const float* __restrict__ emb,
              const int*   __restrict__ labels,
              const float* __restrict__ cent,
              float* __restrict__ ws) {
    __shared__ float cent_s[KCL * DDIM];          // 32 KB centroid stage
    __shared__ float c2_s[KCL];                   // ||c||^2
    __shared__ int   lbl_s[ROWS_PER_BLOCK];
    __shared__ float x2_s[ROWS_PER_BLOCK];        // ||x||^2 per row
    __shared__ float attr_s[KCL];
    __shared__ float reps_s[KCL];
    __shared__ float cnt_s[KCL];

    const int tid        = threadIdx.x;
    const int block_row0 = blockIdx.x * ROWS_PER_BLOCK;

    // ---- stage centroids into LDS (coalesced float4), zero accumulators ----
    {
        const float4* c4  = (const float4*)cent;
        float4*       cs4 = (float4*)cent_s;
        #pragma unroll
        for (int i = 0; i < (KCL * DDIM / 4) / BLOCK; ++i)
            cs4[tid + i * BLOCK] = c4[tid + i * BLOCK];
    }
    if (tid < ROWS_PER_BLOCK) {
        lbl_s[tid] = labels[block_row0 + tid];
        x2_s[tid]  = 0.0f;
    }
    if (tid < KCL) { attr_s[tid] = 0.0f; reps_s[tid] = 0.0f; cnt_s[tid] = 0.0f; }
    __syncthreads();

    // ---- ||c||^2 from the staged tile (tiny, one thread per cluster) ----
    if (tid < KCL) {
        const float4* row = (const float4*)(cent_s + tid * DDIM);
        float s = 0.0f;
        #pragma unroll
        for (int i = 0; i < DDIM / 4; ++i) {
            float4 v = row[i];
            s = fmaf(v.x, v.x, s); s = fmaf(v.y, v.y, s);
            s = fmaf(v.z, v.z, s); s = fmaf(v.w, v.w, s);
        }
        c2_s[tid] = s;
    }

    // ---- WMMA fragment coordinates (wave32) ----
    const int wv   = tid >> 5;
    const int lane = tid & 31;
    const int r    = lane & 15;       // matrix row/col index within half-wave
    const int hi   = lane >> 4;       // half-wave select (K-pair 0/1 vs 2/3)
    const int row0 = block_row0 + wv * 16;

    const float* aptr = emb + (size_t)(row0 + r) * DDIM + hi * 2;

    v8f acc[4] = {v8f{}, v8f{}, v8f{}, v8f{}};
    float px2 = 0.0f;

    // ---- D-loop: 32 steps of 16x16x4 f32 WMMA against 4 cluster tiles ----
    #pragma unroll 4
    for (int d = 0; d < DDIM; d += 4) {
        v2f a = *(const v2f*)(aptr + d);           // A frag: emb[row][d..d+1 | d+2..d+3]
        px2 = fmaf(a.x, a.x, px2);
        px2 = fmaf(a.y, a.y, px2);

        v2f b0 = *(const v2f*)(cent_s + ( 0 + r) * DDIM + d + hi * 2);
        v2f b1 = *(const v2f*)(cent_s + (16 + r) * DDIM + d + hi * 2);
        v2f b2 = *(const v2f*)(cent_s + (32 + r) * DDIM + d + hi * 2);
        v2f b3 = *(const v2f*)(cent_s + (48 + r) * DDIM + d + hi * 2);

        acc[0] = __builtin_amdgcn_wmma_f32_16x16x4_f32(false, a, false, b0,
                                                       (short)0, acc[0], false, false);
        acc[1] = __builtin_amdgcn_wmma_f32_16x16x4_f32(false, a, false, b1,
                                                       (short)0, acc[1], false, false);
        acc[2] = __builtin_amdgcn_wmma_f32_16x16x4_f32(false, a, false, b2,
                                                       (short)0, acc[2], false, false);
        acc[3] = __builtin_amdgcn_wmma_f32_16x16x4_f32(false, a, false, b3,
                                                       (short)0, acc[3], false, false);
    }

    // ---- combine the two half-row partials of ||x||^2 (lanes r and r+16) ----
    atomicAdd(&x2_s[wv * 16 + r], px2);
    __syncthreads();

    // ---- epilogue: distances, per-row reductions, per-cluster bins ----
    const float invKm1 = 1.0f / (float)(KCL - 1);
    #pragma unroll
    for (int j = 0; j < 8; ++j) {
        const int   m    = hi * 8 + j;                 // C layout: lanes>=16 hold M=j+8
        const int   lbl  = lbl_s[wv * 16 + m];
        const float x2v  = x2_s[wv * 16 + m];

        float repSum = 0.0f, ownRep = 0.0f, ownD2 = 0.0f;
        #pragma unroll
        for (int nt = 0; nt < 4; ++nt) {
            const int   n    = nt * 16 + r;            // C layout: N = lane (mod 16)
            const float dot  = acc[nt][j];
            float d2   = fmaxf(fmaf(-2.0f, dot, x2v + c2_s[n]), 0.0f);
            float dist = sqrtf(d2);
            float t    = MARGIN - dist;
            float repv = t * t;                        // relu((m-d)^2) == (m-d)^2
            repSum += repv;
            if (n == lbl) { ownRep = repv; ownD2 = dist * dist; }
        }
        // butterfly sum across the 16-lane half (masks < 16 stay in-half)
        #pragma unroll
        for (int mk = 1; mk < 16; mk <<= 1) {
            repSum += __shfl_xor(repSum, mk, 32);
            ownRep += __shfl_xor(ownRep, mk, 32);
            ownD2  += __shfl_xor(ownD2,  mk, 32);
        }
        if (r == j) {                                  // one lane per row (each half)
            atomicAdd(&attr_s[lbl], ownD2);
            atomicAdd(&reps_s[lbl], (repSum - ownRep) * invKm1);
            atomicAdd(&cnt_s[lbl],  1.0f);
        }
    }
    __syncthreads();

    // ---- flush block-local bins: 192 global atomics per block ----
    if (tid < KCL) {
        atomicAdd(&ws[tid],            attr_s[tid]);
        atomicAdd(&ws[KCL + tid],      reps_s[tid]);
        atomicAdd(&ws[2 * KCL + tid],  cnt_s[tid]);
    }
}

// ---------------------------------------------------------------------------
// Kernel 2: finalize scalar loss
// ---------------------------------------------------------------------------
__global__ void dcl_finalize(const float* __restrict__ ws, float* __restrict__ out) {
    __shared__ float terms[KCL];
    int t = threadIdx.x;
    if (t < KCL) {
        float cnt  = ws[2 * KCL + t];
        float safe = fmaxf(cnt, 1.0f);
        terms[t]   = (cnt > 0.0f) ? (ws[t] + ws[KCL + t]) / safe : 0.0f;
    }
    __syncthreads();
    if (t == 0) {
        float s = 0.0f;
        for (int i = 0; i < KCL; ++i) s += terms[i];
        out[0] = s / (float)KCL;
    }
}

// ---------------------------------------------------------------------------
extern "C" void kernel_launch(void* const* d_in, const int* in_sizes, int n_in,
                              void* d_out, int out_size, void* d_ws, size_t ws_size,
                              hipStream_t stream) {
    const float* emb    = (const float*)d_in[0];   // [N, 128] f32
    const int*   labels = (const int*)  d_in[1];   // [N] int
    const float* cent   = (const float*)d_in[2];   // [64, 128] f32
    float* out = (float*)d_out;
    float* ws  = (float*)d_ws;                     // attr[64] | rep[64] | cnt[64]

    const int N    = in_sizes[1];                  // 262144
    const int grid = N / ROWS_PER_BLOCK;           // 2048 blocks

    dcl_init<<<1, 256, 0, stream>>>(ws, out);
    dcl_main<<<grid, BLOCK, 0, stream>>>(emb, labels, cent, ws);
    dcl_finalize<<<1, 64, 0, stream>>>(ws, out);
}